// OnlineContrastiveLoss_62191126446258
// MI455X (gfx1250) — compile-verified
//
#include <hip/hip_runtime.h>

typedef __attribute__((ext_vector_type(2))) float v2f;
typedef __attribute__((ext_vector_type(4))) float v4f;
typedef __attribute__((ext_vector_type(8))) float v8f;
typedef __attribute__((ext_vector_type(2))) int   v2i;

#define D_EMB 512
#define MARGIN_F 0.5f

// --- zero the 3 double accumulators in workspace ---
__global__ void ocl_init_acc(double* acc) {
    if (threadIdx.x < 3) acc[threadIdx.x] = 0.0;
}

// --- main kernel: 16 pairs per wave via V_WMMA_F32_16X16X4_F32 diagonal ---
// Fragment shape (A 16x4 and B 4x16, f32): lanes 0-15 hold the first half of K,
// lanes 16-31 the second half. WMMA sums over K, so any K permutation is valid
// as long as A and B use the SAME permutation. We exploit this to use b128
// loads: lanes 0-15 load float4 at k, lanes 16-31 at k+4; register pair {0,1}
// is the fragment for K-set {k,k+1,k+4,k+5}, pair {2,3} for {k+2,k+3,k+6,k+7}.
__global__ __launch_bounds__(256) void ocl_pairdot_wmma(
    const float* __restrict__ emb,
    const int*   __restrict__ pairs,   // [n_pairs, 2] int32
    int n_pairs,
    double*      __restrict__ acc,     // [0]=sum pos_loss  [1]=sum neg_raw  [2]=sum neg_loss
    int is_neg)
{
    const int lane      = threadIdx.x & 31;
    const int wave_id   = blockIdx.x * (blockDim.x >> 5) + (threadIdx.x >> 5);
    const int pair_base = wave_id * 16;
    if (pair_base >= n_pairs) return;          // wave-uniform

    const int m     = lane & 15;               // tile row (A) / col (B) for this lane
    const int khalf = (lane >> 4) << 2;        // 0 for lanes 0-15, 4 for lanes 16-31

    const int p  = min(pair_base + m, n_pairs - 1);
    const v2i pr = *(const v2i*)(pairs + 2 * p);   // {a_idx, b_idx} as one b64 load
    const int ia = pr[0];
    const int ib = pr[1];

    const float* __restrict__ arow = emb + (size_t)ia * D_EMB + khalf;
    const float* __restrict__ brow = emb + (size_t)ib * D_EMB + khalf;

    // Two accumulator chains to break the WMMA->WMMA RAW dependency.
    v8f c0 = {};
    v8f c1 = {};
#pragma unroll 4
    for (int k = 0; k < D_EMB; k += 8) {
        v4f a = *(const v4f*)(arow + k);       // global_load_b128
        v4f b = *(const v4f*)(brow + k);       // global_load_b128
        v2f a_lo = __builtin_shufflevector(a, a, 0, 1);
        v2f a_hi = __builtin_shufflevector(a, a, 2, 3);
        v2f b_lo = __builtin_shufflevector(b, b, 0, 1);
        v2f b_hi = __builtin_shufflevector(b, b, 2, 3);
        // (neg_a, A, neg_b, B, c_mod, C, reuse_a, reuse_b)
        c0 = __builtin_amdgcn_wmma_f32_16x16x4_f32(false, a_lo, false, b_lo,
                                                   (short)0, c0, false, false);
        c1 = __builtin_amdgcn_wmma_f32_16x16x4_f32(false, a_hi, false, b_hi,
                                                   (short)0, c1, false, false);
    }
    v8f c = c0 + c1;

    // Diagonal extraction from 16x16 f32 C/D layout:
    //   lanes 0-15  hold (M=r,   N=lane)    in VGPR r
    //   lanes 16-31 hold (M=r+8, N=lane-16) in VGPR r
    // diag(m,m): lanes 0-7 -> c[lane]; lanes 24-31 -> c[lane-24]
    const int sel = (lane < 8) ? lane : ((lane >= 24) ? (lane - 24) : -1);
    float dot = 0.0f;
    dot = (sel == 0) ? c[0] : dot;
    dot = (sel == 1) ? c[1] : dot;
    dot = (sel == 2) ? c[2] : dot;
    dot = (sel == 3) ? c[3] : dot;
    dot = (sel == 4) ? c[4] : dot;
    dot = (sel == 5) ? c[5] : dot;
    dot = (sel == 6) ? c[6] : dot;
    dot = (sel == 7) ? c[7] : dot;

    const int  my_pair = pair_base + ((lane < 8) ? lane : (lane - 16));
    const bool active  = (sel >= 0) && (my_pair < n_pairs);

    float s0, s1;
    if (!is_neg) {
        s0 = active ? (1.0f - dot) : 0.0f;                      // positive_loss
        s1 = 0.0f;
    } else {
        const float raw = 1.0f - dot;
        s0 = active ? raw : 0.0f;                               // negative_raw
        s1 = active ? fmaxf(MARGIN_F - raw, 0.0f) : 0.0f;       // relu(margin - raw)
    }

    // wave32 butterfly reduction
#pragma unroll
    for (int off = 16; off > 0; off >>= 1) {
        s0 += __shfl_xor(s0, off, 32);
        s1 += __shfl_xor(s1, off, 32);
    }

    if (lane == 0) {
        if (!is_neg) {
            atomicAdd(&acc[0], (double)s0);
        } else {
            atomicAdd(&acc[1], (double)s0);
            atomicAdd(&acc[2], (double)s1);
        }
    }
}

// --- finalize: [loss, ap, an] ---
__global__ void ocl_finalize(const double* __restrict__ acc,
                             float* __restrict__ out, int kpos, int kneg) {
    if (threadIdx.x == 0) {
        const double pos_loss = acc[0];
        const double neg_raw  = acc[1];
        const double neg_loss = acc[2];
        out[0] = (float)((pos_loss + neg_loss) / (double)(kpos + kneg));
        out[1] = (float)(pos_loss / (double)kpos);
        out[2] = (float)(neg_raw / (double)kneg);
    }
}

extern "C" void kernel_launch(void* const* d_in, const int* in_sizes, int n_in,
                              void* d_out, int out_size, void* d_ws, size_t ws_size,
                              hipStream_t stream) {
    const float* emb       = (const float*)d_in[0];
    const int*   pos_pairs = (const int*)d_in[1];
    const int*   neg_pairs = (const int*)d_in[2];
    const int    kpos      = in_sizes[1] / 2;
    const int    kneg      = in_sizes[2] / 2;

    double* acc = (double*)d_ws;
    float*  out = (float*)d_out;

    ocl_init_acc<<<1, 32, 0, stream>>>(acc);

    const int waves_pos  = (kpos + 15) / 16;
    const int blocks_pos = (waves_pos + 7) / 8;   // 8 waves (256 threads) per block
    ocl_pairdot_wmma<<<blocks_pos, 256, 0, stream>>>(emb, pos_pairs, kpos, acc, 0);

    const int waves_neg  = (kneg + 15) / 16;
    const int blocks_neg = (waves_neg + 7) / 8;
    ocl_pairdot_wmma<<<blocks_neg, 256, 0, stream>>>(emb, neg_pairs, kneg, acc, 1);

    ocl_finalize<<<1, 32, 0, stream>>>(acc, out, kpos, kneg);
}